// RINModel_11965778886787
// MI455X (gfx1250) — compile-verified
//
#include <hip/hip_runtime.h>
#include <math.h>

// ---------------------------------------------------------------------------
// Model constants (match reference)
// ---------------------------------------------------------------------------
#define LUT_N   4096
#define LUT_EXT 5121          // sin(i*step) for i in [0, 5120]: covers i0+1 and i0+1025
#define SEQ     512
#define NPOS    1024          // B*S
#define DM      128
#define NH      256
#define KDIM    256           // GEMM reduction dim (both mixer and projection)
#define VOCAB   50257
#define PHI_F   1.61803398875f
#define LUT_SCALE 651.89864690f   // LUT_N / (2*pi)

typedef float v2f __attribute__((ext_vector_type(2)));
typedef float v8f __attribute__((ext_vector_type(8)));

#define WMMA_F32(a, b, c) \
  __builtin_amdgcn_wmma_f32_16x16x4_f32(false, (a), false, (b), (short)0, (c), false, false)

// ---------------------------------------------------------------------------
// LUT-interpolated sin/cos, table resident in LDS.
// table[i] = sin(i * 2pi/4096); cos(x) = sin(x + pi/2) -> +1024 entries.
// ---------------------------------------------------------------------------
__device__ __forceinline__ void lut_sc(const float* __restrict__ lut,
                                       float theta, float& s, float& c) {
  float u    = theta * LUT_SCALE;
  float pos  = u - floorf(u * (1.0f / (float)LUT_N)) * (float)LUT_N; // in [0,4096]
  float base = floorf(pos);
  float frac = pos - base;
  int   i0   = ((int)base) & (LUT_N - 1);
  float s0 = lut[i0],        s1 = lut[i0 + 1];
  float c0 = lut[i0 + 1024], c1 = lut[i0 + 1025];
  s = fmaf(frac, s1 - s0, s0);
  c = fmaf(frac, c1 - c0, c0);
}

// ---------------------------------------------------------------------------
// Build extended sin table in workspace (once per launch).
// ---------------------------------------------------------------------------
__global__ void k_lut(float* __restrict__ lut) {
  int i = blockIdx.x * blockDim.x + threadIdx.x;
  if (i < LUT_EXT) lut[i] = sinf((float)i * (float)(6.283185307179586 / LUT_N));
}

// ---------------------------------------------------------------------------
// Precompute wavelength = 1/(1+|W|), biasN = B + PHI*|W| (removes rcp from hot loop)
// ---------------------------------------------------------------------------
__global__ void k_prep(const float* __restrict__ W, const float* __restrict__ B,
                       float* __restrict__ wl, float* __restrict__ bs, int n) {
  int i = blockIdx.x * blockDim.x + threadIdx.x;
  if (i < n) {
    float aw = fabsf(W[i]);
    wl[i] = 1.0f / (1.0f + aw);
    bs[i] = fmaf(PHI_F, aw, B[i]);
  }
}

// ---------------------------------------------------------------------------
// Embedding -> initial complex state hcat[n][0:128]=h_real, [128:256]=h_imag
// One block (128 threads) per position.
// ---------------------------------------------------------------------------
__global__ __launch_bounds__(128) void k_embed(const int* __restrict__ tokens,
                                               const float* __restrict__ emb,
                                               const float* __restrict__ lut_g,
                                               float* __restrict__ hcat) {
  __shared__ float lut[LUT_EXT];
  for (int i = threadIdx.x; i < LUT_EXT; i += 128) lut[i] = lut_g[i];
  __syncthreads();
  const int n = blockIdx.x;                       // 0..1023
  const int d = threadIdx.x;                      // 0..127
  const int tok = tokens[n];
  const float* row = emb + (size_t)tok * (4 * DM);
  float t = (float)((n & (SEQ - 1)) + 1) * PHI_F;
  float s, c;
  lut_sc(lut, fmaf(row[d], t, row[DM + d]), s, c);              // w1*t + b1 -> cos
  hcat[(size_t)n * NH + d] = c;
  lut_sc(lut, fmaf(row[2 * DM + d], t, row[3 * DM + d]), s, c); // w2*t + b2 -> sin
  hcat[(size_t)n * NH + DM + d] = s;
}

// ---------------------------------------------------------------------------
// Fused resonant layer: for a 16(n) x 16(j) tile, loop over din.
// Wave layout: lanes = (2 j) x (16 n) -> parameter loads broadcast, drive in
// LDS with padded stride 257 (conflict-free). theta never materialized.
// ---------------------------------------------------------------------------
__global__ __launch_bounds__(256) void k_resonant(
    const float* __restrict__ lut_g,
    const float* __restrict__ xr, const float* __restrict__ xi, int xstride,
    const float* __restrict__ amp, const float* __restrict__ ang,
    const float* __restrict__ wl,  const float* __restrict__ bs,
    const float* __restrict__ ac,  const float* __restrict__ asn,
    float* __restrict__ outr, float* __restrict__ outi, int ostride,
    int din, float scale, int accumulate) {
  __shared__ float lut[LUT_EXT];
  __shared__ float drive[16 * 257];
  const int tid = threadIdx.x;
  for (int i = tid; i < LUT_EXT; i += 256) lut[i] = lut_g[i];
  __syncthreads();

  const int n0 = blockIdx.x * 16;
  const int j0 = blockIdx.y * 16;

  // drive[n][d] = exp(amp[d]) * (xr*cos(ang[d]+t_n) + xi*sin(ang[d]+t_n))
  for (int idx = tid; idx < 16 * din; idx += 256) {
    int n = idx / din, d = idx - n * din;
    int ng = n0 + n;
    float t = (float)((ng & (SEQ - 1)) + 1) * PHI_F;
    float s, c;
    lut_sc(lut, ang[d] + t, s, c);
    float A = expf(amp[d]);
    float xrv = xr[(size_t)ng * xstride + d];
    float xiv = xi[(size_t)ng * xstride + d];
    drive[n * 257 + d] = A * fmaf(xrv, c, xiv * s);
  }
  __syncthreads();

  const int lj = tid >> 4, ln = tid & 15;
  const int jg = j0 + lj;
  const int ng = n0 + ln;
  const float* __restrict__ wlr = wl  + (size_t)jg * din;
  const float* __restrict__ bsr = bs  + (size_t)jg * din;
  const float* __restrict__ acr = ac  + (size_t)jg * din;
  const float* __restrict__ asr = asn + (size_t)jg * din;
  const float* dr = drive + ln * 257;

  float accC = 0.f, accS = 0.f;
#pragma unroll 4
  for (int d = 0; d < din; ++d) {
    float theta = fmaf(dr[d], wlr[d], bsr[d]);
    float s, c;
    lut_sc(lut, theta, s, c);
    accC = fmaf(c, acr[d], accC);
    accS = fmaf(s, asr[d], accS);
  }
  accC *= scale; accS *= scale;
  size_t o = (size_t)ng * ostride + jg;
  if (accumulate) { outr[o] += accC; outi[o] += accS; }
  else            { outr[o]  = accC; outi[o]  = accS; }
}

// ---------------------------------------------------------------------------
// Fused complex mixer via fp32 WMMA:
//   Or = A1*B1^T - A2*B2^T ; Oi = A1*B2^T + A2*B1^T
// Four wmma chains share four fragment streams (halves load traffic vs two
// separate GEMMs). Software-pipelined one stage ahead so fragment loads for
// step k+4 are in flight while step k's wmmas execute.
// Block: 256 threads (8 waves), tile 16 rows x 128 cols; A tiles in LDS.
// ---------------------------------------------------------------------------
__global__ __launch_bounds__(256) void k_mixer(
    const float* __restrict__ A1, const float* __restrict__ A2,
    const float* __restrict__ B1, const float* __restrict__ B2,
    float* __restrict__ Or, float* __restrict__ Oi, int M) {
  __shared__ float a1t[16 * 257];
  __shared__ float a2t[16 * 257];
  const int tid  = threadIdx.x;
  const int row0 = blockIdx.x * 16;
  const int col0 = blockIdx.y * 128;
  for (int idx = tid; idx < 16 * KDIM; idx += 256) {
    int r = idx >> 8, c = idx & (KDIM - 1);
    a1t[r * 257 + c] = A1[(size_t)(row0 + r) * KDIM + c];
    a2t[r * 257 + c] = A2[(size_t)(row0 + r) * KDIM + c];
  }
  __syncthreads();

  const int lane = tid & 31;
  const int wave = tid >> 5;
  const int m    = lane & 15;       // A row / B column within tile
  const int hi   = lane >> 4;       // K-half select (offsets 0,1 vs 2,3)
  const int col  = col0 + wave * 16 + m;
  const float* arow1 = a1t + m * 257;
  const float* arow2 = a2t + m * 257;
  const float* b1p = B1 + (size_t)col * KDIM;
  const float* b2p = B2 + (size_t)col * KDIM;

  v8f c11 = {}, c22 = {}, c12 = {}, c21 = {};
  v2f a1c, a2c, b1c, b2c;
  {
    int kb = hi * 2;
    a1c.x = arow1[kb]; a1c.y = arow1[kb + 1];
    a2c.x = arow2[kb]; a2c.y = arow2[kb + 1];
    b1c.x = b1p[kb];   b1c.y = b1p[kb + 1];
    b2c.x = b2p[kb];   b2c.y = b2p[kb + 1];
  }
#pragma unroll 4
  for (int kk = 0; kk < KDIM - 4; kk += 4) {
    const int kb = kk + 4 + hi * 2;
    v2f a1n, a2n, b1n, b2n;
    a1n.x = arow1[kb]; a1n.y = arow1[kb + 1];
    a2n.x = arow2[kb]; a2n.y = arow2[kb + 1];
    b1n.x = b1p[kb];   b1n.y = b1p[kb + 1];
    b2n.x = b2p[kb];   b2n.y = b2p[kb + 1];
    c11 = WMMA_F32(a1c, b1c, c11);
    c22 = WMMA_F32(a2c, b2c, c22);
    c12 = WMMA_F32(a1c, b2c, c12);
    c21 = WMMA_F32(a2c, b1c, c21);
    a1c = a1n; a2c = a2n; b1c = b1n; b2c = b2n;
  }
  c11 = WMMA_F32(a1c, b1c, c11);
  c22 = WMMA_F32(a2c, b2c, c22);
  c12 = WMMA_F32(a1c, b2c, c12);
  c21 = WMMA_F32(a2c, b1c, c21);

  const int rbase = row0 + hi * 8;
#pragma unroll
  for (int r = 0; r < 8; ++r) {
    Or[(size_t)(rbase + r) * M + col] = c11[r] - c22[r];
    Oi[(size_t)(rbase + r) * M + col] = c12[r] + c21[r];
  }
}

// ---------------------------------------------------------------------------
// Output projection: logits = hcat @ out_w^T + out_b  (M = VOCAB, ragged edge)
// Column index clamped for loads (EXEC stays all-1 through WMMA), store masked.
// Software-pipelined two stages ahead (loads for k+8 in flight during wmma k).
// ---------------------------------------------------------------------------
__global__ __launch_bounds__(256) void k_gemm_proj(
    const float* __restrict__ A, const float* __restrict__ B,
    const float* __restrict__ bias, float* __restrict__ Out, int M) {
  __shared__ float at[16 * 257];
  const int tid  = threadIdx.x;
  const int row0 = blockIdx.x * 16;
  const int col0 = blockIdx.y * 128;
  for (int idx = tid; idx < 16 * KDIM; idx += 256) {
    int r = idx >> 8, c = idx & (KDIM - 1);
    at[r * 257 + c] = A[(size_t)(row0 + r) * KDIM + c];
  }
  __syncthreads();

  const int lane = tid & 31;
  const int wave = tid >> 5;
  const int m    = lane & 15;
  const int hi   = lane >> 4;
  const int col  = col0 + wave * 16 + m;
  const int colc = col < M ? col : M - 1;    // clamp: keep EXEC full for WMMA
  const float* arow = at + m * 257;
  const float* bp   = B + (size_t)colc * KDIM;
  const float  bv   = bias[colc];

  v8f acc = {};
  v2f a0, b0, a1, b1;
  {
    int kb = hi * 2;
    a0.x = arow[kb];     a0.y = arow[kb + 1];
    b0.x = bp[kb];       b0.y = bp[kb + 1];
    a1.x = arow[kb + 4]; a1.y = arow[kb + 5];
    b1.x = bp[kb + 4];   b1.y = bp[kb + 5];
  }
#pragma unroll 4
  for (int kk = 0; kk < KDIM - 8; kk += 4) {
    const int kb = kk + 8 + hi * 2;
    v2f an, bn;
    an.x = arow[kb]; an.y = arow[kb + 1];
    bn.x = bp[kb];   bn.y = bp[kb + 1];
    acc = WMMA_F32(a0, b0, acc);
    a0 = a1; b0 = b1; a1 = an; b1 = bn;
  }
  acc = WMMA_F32(a0, b0, acc);
  acc = WMMA_F32(a1, b1, acc);

  if (col < M) {
    const int rbase = row0 + hi * 8;
#pragma unroll
    for (int r = 0; r < 8; ++r)
      Out[(size_t)(rbase + r) * M + col] = acc[r] + bv;
  }
}

// ---------------------------------------------------------------------------
// Host launcher.
// d_in flat order: tokens, embedding, then per layer:
//   up{amp,ang,W,B,attn_cos,attn_sin}, mixer_wr, mixer_wi,
//   down{amp,ang,W,B,attn_cos,attn_sin}, then out_w, out_b.
// ---------------------------------------------------------------------------
extern "C" void kernel_launch(void* const* d_in, const int* in_sizes, int n_in,
                              void* d_out, int out_size, void* d_ws, size_t ws_size,
                              hipStream_t stream) {
  (void)in_sizes; (void)n_in; (void)out_size; (void)ws_size;
  const int*   tokens    = (const int*)d_in[0];
  const float* embedding = (const float*)d_in[1];
  const float* out_w     = (const float*)d_in[30];
  const float* out_b     = (const float*)d_in[31];
  float* logits = (float*)d_out;

  float* ws   = (float*)d_ws;
  float* lut  = ws;                       // 5124 floats
  float* hcat = ws + 5124;                // NPOS*NH
  float* ur   = hcat + NPOS * NH;
  float* ui   = ur   + NPOS * NH;
  float* mr   = ui   + NPOS * NH;
  float* mi   = mr   + NPOS * NH;
  float* prep = mi   + NPOS * NH;         // per layer: up_wl, up_bs, dn_wl, dn_bs (32768 each)

  k_lut<<<dim3((LUT_EXT + 255) / 256), dim3(256), 0, stream>>>(lut);

  // Precompute wavelength/bias for all resonant param sets
  for (int l = 0; l < 2; ++l) {
    const int base = 2 + l * 14;
    const float* upW = (const float*)d_in[base + 2];
    const float* upB = (const float*)d_in[base + 3];
    const float* dnW = (const float*)d_in[base + 10];
    const float* dnB = (const float*)d_in[base + 11];
    float* up_wl = prep + (size_t)l * 131072;
    float* up_bs = up_wl + 32768;
    float* dn_wl = up_bs + 32768;
    float* dn_bs = dn_wl + 32768;
    k_prep<<<dim3(128), dim3(256), 0, stream>>>(upW, upB, up_wl, up_bs, NH * DM);
    k_prep<<<dim3(128), dim3(256), 0, stream>>>(dnW, dnB, dn_wl, dn_bs, DM * NH);
  }

  k_embed<<<dim3(NPOS), dim3(128), 0, stream>>>(tokens, embedding, lut, hcat);

  const float scale_up = 0.088388347648318447f;   // 1/sqrt(128)
  const float scale_dn = 0.0625f;                 // 1/sqrt(256)

  for (int l = 0; l < 2; ++l) {
    const int base = 2 + l * 14;
    const float* up_amp = (const float*)d_in[base + 0];
    const float* up_ang = (const float*)d_in[base + 1];
    const float* up_ac  = (const float*)d_in[base + 4];
    const float* up_as  = (const float*)d_in[base + 5];
    const float* wr_    = (const float*)d_in[base + 6];
    const float* wi_    = (const float*)d_in[base + 7];
    const float* dn_amp = (const float*)d_in[base + 8];
    const float* dn_ang = (const float*)d_in[base + 9];
    const float* dn_ac  = (const float*)d_in[base + 12];
    const float* dn_as  = (const float*)d_in[base + 13];
    float* up_wl = prep + (size_t)l * 131072;
    float* up_bs = up_wl + 32768;
    float* dn_wl = up_bs + 32768;
    float* dn_bs = dn_wl + 32768;

    // up: (xr,xi) = hcat halves -> ur, ui   (din=128, nout=256)
    k_resonant<<<dim3(NPOS / 16, NH / 16), dim3(256), 0, stream>>>(
        lut, hcat, hcat + DM, NH, up_amp, up_ang, up_wl, up_bs, up_ac, up_as,
        ur, ui, NH, DM, scale_up, 0);

    // fused mixer: mr = ur*wr^T - ui*wi^T ; mi = ur*wi^T + ui*wr^T
    k_mixer<<<dim3(NPOS / 16, NH / 128), dim3(256), 0, stream>>>(
        ur, ui, wr_, wi_, mr, mi, NH);

    // down: (xr,xi) = (mr,mi) -> accumulate into hcat (din=256, nout=128)
    k_resonant<<<dim3(NPOS / 16, DM / 16), dim3(256), 0, stream>>>(
        lut, mr, mi, NH, dn_amp, dn_ang, dn_wl, dn_bs, dn_ac, dn_as,
        hcat, hcat + DM, NH, NH, scale_dn, 1);
  }

  // logits = hcat @ out_w^T + out_b
  k_gemm_proj<<<dim3(NPOS / 16, (VOCAB + 127) / 128), dim3(256), 0, stream>>>(
      hcat, out_w, out_b, logits, VOCAB);
}